// ConditionalDecoder_31379031065027
// MI455X (gfx1250) — compile-verified
//
#include <hip/hip_runtime.h>
#include <hip/hip_bf16.h>

typedef __attribute__((ext_vector_type(2))) float v2f;
typedef __attribute__((ext_vector_type(8))) float v8f;

#define STG 68          // LDS stage row stride (floats): bank-conflict free, 8B aligned
#define XSTG 36         // input tile stride (33 padded to 36 = 9 K-steps of 4)
#define WAVES 4         // waves per block in decode kernel

__device__ __forceinline__ v8f wmma_f32(v2f a, v2f b, v8f c) {
    // D = A(16x4 f32) * B(4x16 f32) + C(16x16 f32)
    return __builtin_amdgcn_wmma_f32_16x16x4_f32(false, a, false, b, (short)0, c,
                                                 false, false);
}

struct Params {
    const float *z, *tp;
    const float *lin_w, *lin_b;
    const float *nn1_w, *nn1_b, *nn2_w, *nn2_b;
    const float *nd1_w, *nd1_b, *nd_g, *nd_be, *nd_rm, *nd_rv, *nd2_w, *nd2_b;
    const float *pd1_w, *pd1_b, *pd_g, *pd_be, *pd_rm, *pd_rv, *pd2_w, *pd2_b;
    const float *node_scale, *node_shift, *pos_scale, *pos_shift;
    float *nf_ws, *pos_ws;
    int *num_ws, *num_out;
};

// One 16x16 output tile, K=64, A from LDS (stride STG), B from global [64,64] row-major
__device__ __forceinline__ v8f tile_acc64(const float* __restrict__ H,
                                          const float* __restrict__ W,
                                          float bv, int m, int kh, int n) {
    v8f c;
#pragma unroll
    for (int j = 0; j < 8; ++j) c[j] = bv;
#pragma unroll
    for (int k0 = 0; k0 < 64; k0 += 4) {
        const int k = k0 + kh;
        v2f a = *(const v2f*)(H + m * STG + k);
        v2f b = { W[k * 64 + n], W[(k + 1) * 64 + n] };
        c = wmma_f32(a, b, c);
    }
    return c;
}

__global__ __launch_bounds__(32 * WAVES)
void decode_kernel(Params p) {
    __shared__ __align__(16) float xt[WAVES][16 * XSTG];
    __shared__ __align__(16) float hS[WAVES][16 * STG];
    __shared__ __align__(16) float aS[WAVES][16 * STG];

    const int wave = threadIdx.x >> 5;
    const int lane = threadIdx.x & 31;
    const int m    = lane & 15;           // row (A) / col (B,D) within tile
    const int kh   = (lane >> 4) << 1;    // K sub-offset for this half-wave: 0 or 2
    const int Mb   = (lane >> 4) << 3;    // D-fragment M base for this half-wave
    const int r0   = blockIdx.x * (16 * WAVES) + wave * 16;

    float* X = xt[wave];
    float* H = hS[wave];
    float* A = aS[wave];

    // ---- stage input tile [16 x 36]: cols 0..31 = z, 32 = target, 33..35 = 0 ----
    for (int idx = lane; idx < 16 * XSTG; idx += 32) {
        const int row = idx / XSTG, col = idx - row * XSTG;
        float v = 0.f;
        if (col < 32)       v = p.z[(size_t)(r0 + row) * 32 + col];
        else if (col == 32) v = p.tp[r0 + row];
        X[idx] = v;
    }
    __syncthreads();

    // ---- h = X @ lin_w + lin_b   (K = 33, padded to 36) ----
    for (int t = 0; t < 4; ++t) {
        const int n = t * 16 + m;
        const float bv = p.lin_b[n];
        v8f c;
#pragma unroll
        for (int j = 0; j < 8; ++j) c[j] = bv;
#pragma unroll
        for (int k0 = 0; k0 < 36; k0 += 4) {
            const int k = k0 + kh;
            v2f a = *(const v2f*)(X + m * XSTG + k);
            v2f b;
            b.x = (k     < 33) ? p.lin_w[k * 64 + n]       : 0.f;
            b.y = (k + 1 < 33) ? p.lin_w[(k + 1) * 64 + n] : 0.f;
            c = wmma_f32(a, b, c);
        }
#pragma unroll
        for (int j = 0; j < 8; ++j) H[(Mb + j) * STG + n] = c[j];
    }
    __syncthreads();

    // ---- num-nodes head: a1 = relu(h @ nn1_w + nn1_b) ----
    for (int t = 0; t < 4; ++t) {
        const int n = t * 16 + m;
        v8f c = tile_acc64(H, p.nn1_w, p.nn1_b[n], m, kh, n);
#pragma unroll
        for (int j = 0; j < 8; ++j) A[(Mb + j) * STG + n] = fmaxf(c[j], 0.f);
    }
    __syncthreads();

    if (lane < 16) {  // logit -> sigmoid -> int node count
        float s = p.nn2_b[0];
#pragma unroll 4
        for (int k = 0; k < 64; ++k) s += A[m * STG + k] * p.nn2_w[k];
        const float sig = 1.f / (1.f + expf(-s));
        const int nn = (int)(sig * 30.f + 5.f);
        p.num_ws[r0 + m]  = nn;
        p.num_out[r0 + m] = nn;
    }
    __syncthreads();

    // ---- node-feature head: BN(relu(h @ nd1_w + nd1_b)) (BN fused per-column) ----
    for (int t = 0; t < 4; ++t) {
        const int n = t * 16 + m;
        v8f c = tile_acc64(H, p.nd1_w, p.nd1_b[n], m, kh, n);
        const float sc = p.nd_g[n] * rsqrtf(p.nd_rv[n] + 1e-5f);
        const float rm = p.nd_rm[n], be = p.nd_be[n];
#pragma unroll
        for (int j = 0; j < 8; ++j)
            A[(Mb + j) * STG + n] = (fmaxf(c[j], 0.f) - rm) * sc + be;
    }
    __syncthreads();

    {
        const float ns = *p.node_scale, nsh = *p.node_shift;
        if (lane < 16) {
            for (int n = 0; n < 11; ++n) {
                float s = p.nd2_b[n];
#pragma unroll 4
                for (int k = 0; k < 64; ++k) s += A[m * STG + k] * p.nd2_w[k * 11 + n];
                p.nf_ws[(size_t)(r0 + m) * 11 + n] = s * ns + nsh;
            }
        }
    }
    __syncthreads();

    // ---- position head ----
    for (int t = 0; t < 4; ++t) {
        const int n = t * 16 + m;
        v8f c = tile_acc64(H, p.pd1_w, p.pd1_b[n], m, kh, n);
        const float sc = p.pd_g[n] * rsqrtf(p.pd_rv[n] + 1e-5f);
        const float rm = p.pd_rm[n], be = p.pd_be[n];
#pragma unroll
        for (int j = 0; j < 8; ++j)
            A[(Mb + j) * STG + n] = (fmaxf(c[j], 0.f) - rm) * sc + be;
    }
    __syncthreads();

    {
        const float ps = *p.pos_scale, psh = *p.pos_shift;
        if (lane < 16) {
            for (int n = 0; n < 3; ++n) {
                float s = p.pd2_b[n];
#pragma unroll 4
                for (int k = 0; k < 64; ++k) s += A[m * STG + k] * p.pd2_w[k * 3 + n];
                p.pos_ws[(size_t)(r0 + m) * 3 + n] = s * ps + psh;
            }
        }
    }
}

// ---------------- exclusive scan of num_nodes (two level) ----------------
__global__ void scan_block(const int* __restrict__ num, int* __restrict__ offs,
                           int* __restrict__ blockSums) {
    __shared__ int sh[256];
    const int tid = threadIdx.x;
    const int i = blockIdx.x * 256 + tid;
    const int v = num[i];
    int x = v;
    sh[tid] = x;
    __syncthreads();
    for (int d = 1; d < 256; d <<= 1) {
        int y = (tid >= d) ? sh[tid - d] : 0;
        __syncthreads();
        x += y;
        sh[tid] = x;
        __syncthreads();
    }
    offs[i] = x - v;  // exclusive, block-local
    if (tid == 255) blockSums[blockIdx.x] = x;
}

__global__ void scan_mid(const int* __restrict__ blockSums, int* __restrict__ blockBase,
                         int* __restrict__ total, int nb) {
    __shared__ int sh[512];
    const int tid = threadIdx.x;
    const int v = (tid < nb) ? blockSums[tid] : 0;
    int x = v;
    sh[tid] = x;
    __syncthreads();
    for (int d = 1; d < 512; d <<= 1) {
        int y = (tid >= d) ? sh[tid - d] : 0;
        __syncthreads();
        x += y;
        sh[tid] = x;
        __syncthreads();
    }
    if (tid < nb) blockBase[tid] = x - v;
    if (tid == 511) *total = x;
}

__global__ void add_base(int* __restrict__ offs, const int* __restrict__ blockBase) {
    offs[blockIdx.x * 256 + threadIdx.x] += blockBase[blockIdx.x];
}

// ---------------- ragged repeat: one wave per source row ----------------
__global__ __launch_bounds__(256)
void scatter_kernel(const float* __restrict__ nf, const float* __restrict__ pos,
                    const int* __restrict__ num, const int* __restrict__ offs,
                    float* __restrict__ out_nf, float* __restrict__ out_pos) {
    __shared__ float snf[8][12];
    __shared__ float spo[8][4];
    const int wave = threadIdx.x >> 5, lane = threadIdx.x & 31;
    const size_t i = (size_t)blockIdx.x * 8 + wave;
    if (lane < 11) snf[wave][lane] = nf[i * 11 + lane];
    if (lane < 3)  spo[wave][lane] = pos[i * 3 + lane];
    __syncthreads();
    const int n = num[i];
    const long off = offs[i];
    const int cnt11 = n * 11;
    float* o1 = out_nf + (size_t)off * 11;
    for (int e = lane; e < cnt11; e += 32) o1[e] = snf[wave][e % 11];
    const int cnt3 = n * 3;
    float* o2 = out_pos + (size_t)off * 3;
    for (int e = lane; e < cnt3; e += 32) o2[e] = spo[wave][e % 3];
}

// tail padding: rows [total_sum, B*35) replicate the last source row (clip-gather)
__global__ void fill_kernel(const float* __restrict__ nf, const float* __restrict__ pos,
                            const int* __restrict__ total_p,
                            float* __restrict__ out_nf, float* __restrict__ out_pos,
                            int B, long total) {
    const long j = (long)(*total_p) + (long)blockIdx.x * blockDim.x + threadIdx.x;
    if (j >= total) return;
    const size_t s = (size_t)(B - 1);
    float* o1 = out_nf + (size_t)j * 11;
#pragma unroll
    for (int c = 0; c < 11; ++c) o1[c] = nf[s * 11 + c];
    float* o2 = out_pos + (size_t)j * 3;
#pragma unroll
    for (int c = 0; c < 3; ++c) o2[c] = pos[s * 3 + c];
}

extern "C" void kernel_launch(void* const* d_in, const int* in_sizes, int n_in,
                              void* d_out, int out_size, void* d_ws, size_t ws_size,
                              hipStream_t stream) {
    (void)n_in; (void)out_size; (void)ws_size;
    const int B = in_sizes[0] / 32;          // z is [B,32]
    const long total = (long)B * 35;         // MAX_NODES padded length

    // workspace layout
    float* nf_ws  = (float*)d_ws;                            // B*11 f32
    float* pos_ws = nf_ws + (size_t)B * 11;                  // B*3  f32
    int*   num_ws = (int*)(pos_ws + (size_t)B * 3);          // B    i32
    int*   offs   = num_ws + B;                              // B    i32
    int*   bSums  = offs + B;                                // 512
    int*   bBase  = bSums + 512;                             // 512
    int*   totalp = bBase + 512;                             // 1

    // output layout: nf [total,11] | pos [total,3] | num_nodes [B] (i32)
    float* out_nf  = (float*)d_out;
    float* out_pos = out_nf + (size_t)total * 11;
    int*   out_num = (int*)(out_pos + (size_t)total * 3);

    Params p;
    p.z = (const float*)d_in[0];   p.tp = (const float*)d_in[1];
    p.lin_w = (const float*)d_in[3];  p.lin_b = (const float*)d_in[4];
    p.nn1_w = (const float*)d_in[5];  p.nn1_b = (const float*)d_in[6];
    p.nn2_w = (const float*)d_in[7];  p.nn2_b = (const float*)d_in[8];
    p.nd1_w = (const float*)d_in[9];  p.nd1_b = (const float*)d_in[10];
    p.nd_g  = (const float*)d_in[11]; p.nd_be = (const float*)d_in[12];
    p.nd_rm = (const float*)d_in[13]; p.nd_rv = (const float*)d_in[14];
    p.nd2_w = (const float*)d_in[15]; p.nd2_b = (const float*)d_in[16];
    p.pd1_w = (const float*)d_in[17]; p.pd1_b = (const float*)d_in[18];
    p.pd_g  = (const float*)d_in[19]; p.pd_be = (const float*)d_in[20];
    p.pd_rm = (const float*)d_in[21]; p.pd_rv = (const float*)d_in[22];
    p.pd2_w = (const float*)d_in[23]; p.pd2_b = (const float*)d_in[24];
    p.node_scale = (const float*)d_in[25]; p.node_shift = (const float*)d_in[26];
    p.pos_scale  = (const float*)d_in[27]; p.pos_shift  = (const float*)d_in[28];
    p.nf_ws = nf_ws; p.pos_ws = pos_ws; p.num_ws = num_ws; p.num_out = out_num;

    // 1) fused encoder + heads (WMMA f32)
    decode_kernel<<<B / (16 * WAVES), 32 * WAVES, 0, stream>>>(p);

    // 2-4) exclusive scan of num_nodes
    const int nb = B / 256;  // 512
    scan_block<<<nb, 256, 0, stream>>>(num_ws, offs, bSums);
    scan_mid<<<1, 512, 0, stream>>>(bSums, bBase, totalp, nb);
    add_base<<<nb, 256, 0, stream>>>(offs, bBase);

    // 5) ragged repeat + tail padding
    scatter_kernel<<<B / 8, 256, 0, stream>>>(nf_ws, pos_ws, num_ws, offs, out_nf, out_pos);
    const long tail_max = total - (long)B * 5;  // num_nodes >= 5
    fill_kernel<<<(unsigned)((tail_max + 255) / 256), 256, 0, stream>>>(
        nf_ws, pos_ws, totalp, out_nf, out_pos, B, total);
}